// ContrastiveLoss_51591147159598
// MI455X (gfx1250) — compile-verified
//
#include <hip/hip_runtime.h>
#include <stdint.h>

typedef float v2f __attribute__((ext_vector_type(2)));
typedef float v8f __attribute__((ext_vector_type(8)));

#define B_N   8192
#define D_K   256
#define TILE  128
#define KC    32
#define KPAD  (KC + 4)            // 36 floats: keeps 16B alignment, scrambles banks
#define NT    (B_N / TILE)        // 64 tile rows/cols
#define NBLK  (NT * (NT + 1) / 2) // 2080 upper-triangular tiles
#define MARGIN 2.0f

#define WAIT_ASYNC() asm volatile("s_wait_asynccnt 0x0" ::: "memory")

__global__ void cl_zero_ws(double* ws) { ws[0] = 0.0; }

// Async-DMA one KC-wide chunk of A and B tiles into LDS (no VGPR round trip).
// 256 threads x 4 iters x 2 matrices x 16B/lane = full 128x32 f32 chunk each.
__device__ __forceinline__ void stage_async(const float* __restrict__ gA,
                                            const float* __restrict__ gB,
                                            float (*Asb)[KPAD], float (*Bsb)[KPAD],
                                            int tid)
{
    #pragma unroll
    for (int i = 0; i < 4; ++i) {
        const int c   = tid + 256 * i;   // 0..1023 float4 chunks
        const int row = c >> 3;          // 8 float4 per 32-float row
        const int c4  = (c & 7) << 2;    // float column 0,4,...,28
        const unsigned  la_a = (unsigned)(uintptr_t)&Asb[row][c4];
        const unsigned  la_b = (unsigned)(uintptr_t)&Bsb[row][c4];
        const unsigned long long ga_a =
            (unsigned long long)(uintptr_t)(gA + (size_t)row * D_K + c4);
        const unsigned long long ga_b =
            (unsigned long long)(uintptr_t)(gB + (size_t)row * D_K + c4);
        asm volatile("global_load_async_to_lds_b128 %0, %1, off"
                     :: "v"(la_a), "v"(ga_a) : "memory");
        asm volatile("global_load_async_to_lds_b128 %0, %1, off"
                     :: "v"(la_b), "v"(ga_b) : "memory");
    }
}

// One KC-wide chunk of the 128x128 tile: 8 k-steps x 8 independent WMMAs.
__device__ __forceinline__ void compute_chunk(const float (*Asb)[KPAD],
                                              const float (*Bsb)[KPAD],
                                              int wr, int wc, int l15, int lh,
                                              v8f acc[2][4])
{
    #pragma unroll
    for (int kk = 0; kk < KC; kk += 4) {
        // A fragments: lane l holds M = l&15, K = kk + 2*(l>>4) + {0,1}
        v2f afr[2];
        #pragma unroll
        for (int rm = 0; rm < 2; ++rm)
            afr[rm] = *(const v2f*)&Asb[wr * 32 + rm * 16 + l15][kk + 2 * lh];
        // B fragments: lane l holds N = l&15, same K pairing
        v2f bfr[4];
        #pragma unroll
        for (int cn = 0; cn < 4; ++cn)
            bfr[cn] = *(const v2f*)&Bsb[wc * 64 + cn * 16 + l15][kk + 2 * lh];

        #pragma unroll
        for (int rm = 0; rm < 2; ++rm)
            #pragma unroll
            for (int cn = 0; cn < 4; ++cn)
                acc[rm][cn] = __builtin_amdgcn_wmma_f32_16x16x4_f32(
                    false, afr[rm], false, bfr[cn],
                    (short)0, acc[rm][cn], false, false);
    }
}

__global__ __launch_bounds__(256)
void cl_gemm_loss(const float* __restrict__ emb,
                  const int*   __restrict__ keys,
                  double*      __restrict__ ws)
{
    __shared__ float As[2][TILE][KPAD];   // double-buffered A (36 KB)
    __shared__ float Bs[2][TILE][KPAD];   // double-buffered B (36 KB)
    __shared__ int   krow[TILE];
    __shared__ int   kcol[TILE];
    __shared__ float red[8];

    // ---- decode linear block id -> (bi, bj) with bi <= bj (upper triangle) ----
    int t  = (int)blockIdx.x;
    int bi = 0;
    while (t >= NT - bi) { t -= NT - bi; ++bi; }
    const int bj = bi + t;

    const int rowA0 = bi * TILE;
    const int rowB0 = bj * TILE;
    const float* gA = emb + (size_t)rowA0 * D_K;
    const float* gB = emb + (size_t)rowB0 * D_K;

    const int tid = threadIdx.x;
    const int w   = tid >> 5;
    const int l   = tid & 31;
    const int l15 = l & 15;
    const int lh  = l >> 4;

    const int wr = w >> 1;         // 4 wave-rows x 32 rows
    const int wc = w & 1;          // 2 wave-cols x 64 cols

    v8f acc[2][4];
    #pragma unroll
    for (int rm = 0; rm < 2; ++rm)
        #pragma unroll
        for (int cn = 0; cn < 4; ++cn)
            acc[rm][cn] = (v8f){0.f,0.f,0.f,0.f,0.f,0.f,0.f,0.f};

    // ---- prologue: async-stage chunk 0 into buffer 0 ----
    stage_async(gA, gB, As[0], Bs[0], tid);
    WAIT_ASYNC();
    __syncthreads();

    // ---- K loop, double buffered, unrolled x2 so buffer idx is constant ----
    for (int k0 = 0; k0 < D_K; k0 += 2 * KC) {
        // stage chunk k0+KC into buf1 while computing buf0
        if (k0 + KC < D_K)
            stage_async(gA + k0 + KC, gB + k0 + KC, As[1], Bs[1], tid);
        compute_chunk(As[0], Bs[0], wr, wc, l15, lh, acc);
        WAIT_ASYNC();
        __syncthreads();

        // stage chunk k0+2*KC into buf0 while computing buf1
        if (k0 + 2 * KC < D_K)
            stage_async(gA + k0 + 2 * KC, gB + k0 + 2 * KC, As[0], Bs[0], tid);
        compute_chunk(As[1], Bs[1], wr, wc, l15, lh, acc);
        WAIT_ASYNC();
        __syncthreads();
    }

    // ---------------- stage order_keys for this tile ----------------
    if (tid < TILE)       krow[tid]        = keys[rowA0 + tid];
    else                  kcol[tid - TILE] = keys[rowB0 + tid - TILE];
    __syncthreads();

    // ---------------- fused loss epilogue + reduction ----------------
    // C/D layout: VGPR r, lanes 0-15 -> M = r, N = lane; lanes 16-31 -> M = r+8
    float lsum = 0.f;
    #pragma unroll
    for (int rm = 0; rm < 2; ++rm) {
        #pragma unroll
        for (int cn = 0; cn < 4; ++cn) {
            const int ck = kcol[wc * 64 + cn * 16 + l15];
            const v8f a  = acc[rm][cn];
            #pragma unroll
            for (int r = 0; r < 8; ++r) {
                const int rloc = wr * 32 + rm * 16 + r + 8 * lh;
                const float dist = 1.0f - a[r];
                const float pos  = dist * dist;
                const float nm   = fmaxf(MARGIN - dist, 0.0f);
                const float neg  = nm * nm;
                lsum += (krow[rloc] == ck) ? pos : neg;
            }
        }
    }
    // diagonal tiles: symmetric with zero diagonal -> half weight
    if (bi == bj) lsum *= 0.5f;

    #pragma unroll
    for (int off = 16; off > 0; off >>= 1)
        lsum += __shfl_down(lsum, off, 32);
    if (l == 0) red[w] = lsum;
    __syncthreads();
    if (tid == 0) {
        float bs = 0.f;
        #pragma unroll
        for (int i = 0; i < 8; ++i) bs += red[i];
        atomicAdd(ws, (double)bs);
    }
}

__global__ void cl_finalize(const double* __restrict__ ws, float* __restrict__ out)
{
    const double num_pairs = (double)B_N * (double)(B_N - 1) * 0.5; // 33550336
    out[0] = (float)(ws[0] / num_pairs);
}

extern "C" void kernel_launch(void* const* d_in, const int* in_sizes, int n_in,
                              void* d_out, int out_size, void* d_ws, size_t ws_size,
                              hipStream_t stream)
{
    const float* emb  = (const float*)d_in[0];
    const int*   keys = (const int*)d_in[1];
    double*      acc  = (double*)d_ws;
    float*       out  = (float*)d_out;

    cl_zero_ws<<<1, 1, 0, stream>>>(acc);
    cl_gemm_loss<<<dim3(NBLK), dim3(256), 0, stream>>>(emb, keys, acc);
    cl_finalize<<<1, 1, 0, stream>>>(acc, out);
}